// DecorrelationGradient_67310727463545
// MI455X (gfx1250) — compile-verified
//
#include <hip/hip_runtime.h>
#include <hip/hip_bf16.h>

// DecorrelationGradient on MI455X (gfx1250):
//   out = (1-k)*(G - diag(ms)) + k*(diag(ms) - 1),  G = X^T X / N, ms = diag(G)
//   => off-diag: (1-k)*G_ij - k ; diag: k*G_ii - k
// Core = fp32 SYRK via V_WMMA_F32_16X16X4_F32, 32x64 register block per wave
// (8 WMMAs per k-step from 6 fragment loads). LDS uses a K-pair-interleaved
// layout so every fragment is a single aligned ds_load_b64 (no packing movs);
// row stride ≡ 32 (mod 64) dwords keeps the two half-waves on disjoint banks.
// Split-K with fp32 atomics into d_out, then an elementwise finalize.

#define KAPPA 0.5f
#define D      768
#define NTOT   16384            // 8*2048
#define TILE_M 128              // rows per workgroup (4 waves x 32)
#define TILE_N 64               // cols per workgroup
#define KC     32               // K rows staged in LDS per chunk
#define KP     (KC / 2)         // K pairs per chunk
#define SPLITK 8
#define NTI    (D / TILE_M)     // 6
#define NTJ    (D / TILE_N)     // 12
#define LDA    (2 * TILE_M + 32)  // 288 dwords; mod 64 == 32
#define LDB    (2 * TILE_N + 32)  // 160 dwords; mod 64 == 32

typedef __attribute__((ext_vector_type(2))) float v2f;
typedef __attribute__((ext_vector_type(8))) float v8f;

__global__ __launch_bounds__(256) void decorr_zero_kernel(float* __restrict__ out, int n) {
    int i = blockIdx.x * blockDim.x + threadIdx.x;
    if (i < n) out[i] = 0.0f;
}

__global__ __launch_bounds__(128) void decorr_syrk_kernel(const float* __restrict__ X,
                                                          float* __restrict__ G) {
    // element (k, c) lives at l?[k/2][2*c + (k&1)] -> fragment = aligned v2f
    __shared__ float lA[KP][LDA];   // X rows, cols i0..i0+127, pair-interleaved
    __shared__ float lB[KP][LDB];   // X rows, cols j0..j0+63,  pair-interleaved

    const int wg   = blockIdx.x;
    const int ks   = wg % SPLITK;          // K-slice
    const int tile = wg / SPLITK;
    const int ti   = tile / NTJ;
    const int tj   = tile % NTJ;
    const int i0   = ti * TILE_M;          // output row base
    const int j0   = tj * TILE_N;          // output col base

    const int t    = threadIdx.x;
    const int wave = t >> 5;
    const int lane = t & 31;
    const int wm   = wave * 32;            // wave's 32-row band inside 128
    const int half = lane >> 4;            // 0: K pair (0,1)  1: K pair (2,3)
    const int lc   = lane & 15;

    v8f acc[2][4] = {};                    // 32x64 per wave

    const int kslice = NTOT / SPLITK;      // 2048
    const int k0 = ks * kslice;
    const int k1 = k0 + kslice;

    const int cB = t & 63;                 // lB staging col
    const int pB = t >> 6;                 // lB staging pair-row base (0..1)

    for (int kb = k0; kb < k1; kb += KC) {
        // ---- stage strips of X into LDS, one b64 store per K-pair ----
        #pragma unroll 4
        for (int p = 0; p < KP; ++p) {                    // 128 threads = 1 col each
            v2f v;
            v.x = X[(kb + 2 * p + 0) * D + i0 + t];
            v.y = X[(kb + 2 * p + 1) * D + i0 + t];
            *(v2f*)&lA[p][2 * t] = v;
        }
        #pragma unroll 4
        for (int p = pB; p < KP; p += 2) {                // 2 pair-rows per pass
            v2f v;
            v.x = X[(kb + 2 * p + 0) * D + j0 + cB];
            v.y = X[(kb + 2 * p + 1) * D + j0 + cB];
            *(v2f*)&lB[p][2 * cB] = v;
        }
        // hint-prefetch next chunk (uniform branch; reconverges before WMMA)
        if (kb + KC < k1) {
            int r = t & 31;
            if (t < 32)       __builtin_prefetch(&X[(kb + KC + r) * D + i0], 0, 1);
            else if (t < 64)  __builtin_prefetch(&X[(kb + KC + r) * D + i0 + 64], 0, 1);
            else if (t < 96)  __builtin_prefetch(&X[(kb + KC + r) * D + j0], 0, 1);
        }
        __syncthreads();

        // ---- 8 k-steps of 4; 8 WMMAs from 6 b64 fragment loads per step ----
        #pragma unroll
        for (int kk = 0; kk < KC; kk += 4) {
            const int pr = (kk >> 1) + half;   // this half-wave's pair-row
            const v2f* pa = (const v2f*)&lA[pr][2 * (wm + lc)];
            const v2f* pb = (const v2f*)&lB[pr][2 * lc];
            v2f a[2], b[4];
            a[0] = pa[0];
            a[1] = pa[16];
            #pragma unroll
            for (int n = 0; n < 4; ++n) b[n] = pb[16 * n];
            #pragma unroll
            for (int m = 0; m < 2; ++m)
                #pragma unroll
                for (int n = 0; n < 4; ++n)
                    acc[m][n] = __builtin_amdgcn_wmma_f32_16x16x4_f32(
                        false, a[m], false, b[n], (short)0, acc[m][n], false, false);
        }
        __syncthreads();
    }

    // ---- epilogue: C layout (VGPR r -> row r + 8*half, col = lane%16) ----
    #pragma unroll
    for (int m = 0; m < 2; ++m) {
        const int row0 = i0 + wm + m * 16 + 8 * half;
        #pragma unroll
        for (int n = 0; n < 4; ++n) {
            const int col = j0 + n * 16 + lc;
            #pragma unroll
            for (int r = 0; r < 8; ++r)
                atomicAdd(&G[(row0 + r) * D + col], acc[m][n][r]);
        }
    }
}

__global__ __launch_bounds__(256) void decorr_finalize_kernel(float* __restrict__ out, int n) {
    int idx = blockIdx.x * blockDim.x + threadIdx.x;
    if (idx >= n) return;
    const int i = idx / D;
    const int j = idx % D;
    const float g = out[idx] * (1.0f / (float)NTOT);   // gram = raw sum / N
    // off-diag: (1-k)*g - k ; diag: k*g - k  (ms_i == g_ii cancels algebraically)
    const float v = (i == j) ? (KAPPA * g - KAPPA)
                             : ((1.0f - KAPPA) * g - KAPPA);
    out[idx] = v;
}

extern "C" void kernel_launch(void* const* d_in, const int* in_sizes, int n_in,
                              void* d_out, int out_size, void* d_ws, size_t ws_size,
                              hipStream_t stream) {
    (void)in_sizes; (void)n_in; (void)d_ws; (void)ws_size; (void)out_size;
    const float* X = (const float*)d_in[0];
    float* out = (float*)d_out;

    const int n = D * D;
    decorr_zero_kernel<<<(n + 255) / 256, 256, 0, stream>>>(out, n);
    decorr_syrk_kernel<<<NTI * NTJ * SPLITK, 128, 0, stream>>>(X, out);
    decorr_finalize_kernel<<<(n + 255) / 256, 256, 0, stream>>>(out, n);
}